// ModSIRENSR_10926396801893
// MI455X (gfx1250) — compile-verified
//
#include <hip/hip_runtime.h>
#include <math.h>

#define N2V 4096
#define N3V 16384
#define HID1 129
#define PAD1 132
#define MP1 144
#define HID2 11
#define PAD2 12
#define OMEGA_C 30.0f
#define SPAN_C 1.1963f
#define TAN30_C 0.57735026918962576451f
#define SCALE_C (1.0f/0.56f - 1.0f)

typedef __attribute__((ext_vector_type(2))) float v2f;
typedef __attribute__((ext_vector_type(8))) float v8f;

// ---------------------------------------------------------------------------
// Per-dim min/max of pos_2 (block 0) and pos_3 (block 1).
// ---------------------------------------------------------------------------
__global__ __launch_bounds__(256) void minmax_kernel(const float* __restrict__ pos2,
                                                     const float* __restrict__ pos3,
                                                     float* __restrict__ mm) {
    const float* p = (blockIdx.x == 0) ? pos2 : pos3;
    const int n = (blockIdx.x == 0) ? N2V : N3V;
    __shared__ float smn[3][256];
    __shared__ float smx[3][256];
    const int tid = threadIdx.x;
    float mn[3] = {1e30f, 1e30f, 1e30f};
    float mx[3] = {-1e30f, -1e30f, -1e30f};
    for (int i = tid; i < n; i += 256) {
        #pragma unroll
        for (int d = 0; d < 3; ++d) {
            float v = p[i * 3 + d];
            mn[d] = fminf(mn[d], v);
            mx[d] = fmaxf(mx[d], v);
        }
    }
    #pragma unroll
    for (int d = 0; d < 3; ++d) { smn[d][tid] = mn[d]; smx[d][tid] = mx[d]; }
    __syncthreads();
    for (int s = 128; s > 0; s >>= 1) {
        if (tid < s) {
            #pragma unroll
            for (int d = 0; d < 3; ++d) {
                smn[d][tid] = fminf(smn[d][tid], smn[d][tid + s]);
                smx[d][tid] = fmaxf(smx[d][tid], smx[d][tid + s]);
            }
        }
        __syncthreads();
    }
    if (tid < 3) {
        mm[blockIdx.x * 6 + tid] = smn[tid][0];
        mm[blockIdx.x * 6 + 3 + tid] = smx[tid][0];
    }
}

// ---------------------------------------------------------------------------
// Build padded / normalized / transformed input buffers.
// ---------------------------------------------------------------------------
__global__ __launch_bounds__(256) void prep_kernel(
    const float* __restrict__ x, const float* __restrict__ pos2,
    const float* __restrict__ pos3, const float* __restrict__ mm,
    float* __restrict__ xp, float* __restrict__ pos2p, float* __restrict__ pos3p,
    float* __restrict__ p2plain, float* __restrict__ p2trans) {
    const int i = blockIdx.x * 256 + threadIdx.x;
    if (i < N3V) {
        #pragma unroll
        for (int d = 0; d < 3; ++d) {
            float mn = mm[6 + d], mx = mm[9 + d];
            pos3p[i * 4 + d] = (pos3[i * 3 + d] - mn) / (mx - mn);
        }
        pos3p[i * 4 + 3] = 0.f;
    }
    if (i < N2V) {
        float p0 = pos2[i * 3 + 0], p1 = pos2[i * 3 + 1], p2v = pos2[i * 3 + 2];
        #pragma unroll
        for (int d = 0; d < 3; ++d) {
            float mn = mm[d], mx = mm[3 + d];
            pos2p[i * 4 + d] = (pos2[i * 3 + d] - mn) / (mx - mn);
        }
        pos2p[i * 4 + 3] = 0.f;
        p2plain[i * 4 + 0] = p0; p2plain[i * 4 + 1] = p1;
        p2plain[i * 4 + 2] = p2v; p2plain[i * 4 + 3] = 0.f;
        float s = 1.0f + SCALE_C * (p1 / SPAN_C);
        p2trans[i * 4 + 0] = (p0 - TAN30_C * p1) * s;
        p2trans[i * 4 + 1] = p1 * s;
        p2trans[i * 4 + 2] = p2v * s;
        p2trans[i * 4 + 3] = 0.f;
        #pragma unroll
        for (int d = 0; d < 5; ++d) xp[i * 8 + d] = x[i * 5 + d];
        #pragma unroll
        for (int d = 5; d < 8; ++d) xp[i * 8 + d] = 0.f;
    }
}

// ---------------------------------------------------------------------------
// Zero-pad W[M][K] -> Wp[Mp][Kp] and B[M] -> Bp[Mp] (once per launch).
// ---------------------------------------------------------------------------
__global__ __launch_bounds__(256) void pack_kernel(const float* __restrict__ W,
                                                   const float* __restrict__ B,
                                                   int M, int K, int Kp, int Mp,
                                                   float* __restrict__ Wp,
                                                   float* __restrict__ Bp) {
    const int idx = blockIdx.x * 256 + threadIdx.x;
    if (idx < Mp * Kp) {
        const int m = idx / Kp, k = idx % Kp;
        Wp[idx] = (m < M && k < K) ? W[m * K + k] : 0.f;
    }
    if (idx < Mp) Bp[idx] = (idx < M) ? B[idx] : 0.f;
}

// ---------------------------------------------------------------------------
// Generic WMMA f32 layer: Y = act( alpha_row * X @ Wp^T + Bp )
// All operands zero-padded: the hot loop is two unconditional b64 loads plus
// one V_WMMA_F32_16X16X4_F32 per K-step. Padding columns yield sin(0)=0.
// ---------------------------------------------------------------------------
template <bool HAS_ALPHA, bool USE_SINE>
__global__ __launch_bounds__(32) void wmma_sine_layer(
    const float* __restrict__ X, int ldx,
    const float* __restrict__ Wp, const float* __restrict__ Bp,
    const float* __restrict__ alphas, int layer,
    float* __restrict__ Y, int ldy, int Mout, float omega) {
    const int lane = threadIdx.x;
    const int r = lane & 15;
    const int hi = lane >> 4;
    const int rowBase = blockIdx.x * 16;
    const int colBase = blockIdx.y * 16;
    const int row = rowBase + r;
    const int n = colBase + r;
    float aScale = 1.0f;
    if (HAS_ALPHA) aScale = alphas[row * 4 + layer];
    const float* xrow = X + (size_t)row * ldx + hi * 2;
    const float* wrow = Wp + (size_t)n * ldx + hi * 2;
    v8f c = {0.f, 0.f, 0.f, 0.f, 0.f, 0.f, 0.f, 0.f};
    for (int k0 = 0; k0 < ldx; k0 += 4) {
        v2f a = *(const v2f*)(xrow + k0);
        if (HAS_ALPHA) { a.x *= aScale; a.y *= aScale; }
        v2f b = *(const v2f*)(wrow + k0);
        c = __builtin_amdgcn_wmma_f32_16x16x4_f32(false, a, false, b, (short)0, c,
                                                  false, false);
    }
    #pragma unroll
    for (int v = 0; v < 8; ++v) {
        const int orow = rowBase + v + hi * 8;
        const int ocol = colBase + r;
        if (ocol < Mout) {
            float val = c[v] + Bp[ocol];
            if (USE_SINE) val = sinf(omega * val);
            Y[(size_t)orow * ldy + ocol] = val;
        }
    }
}

// ---------------------------------------------------------------------------
// k=3 inverse-d2 KNN interpolation, transformed + plain metrics, masked select.
// ---------------------------------------------------------------------------
__global__ __launch_bounds__(256) void knn_kernel(
    const float* __restrict__ p2plain, const float* __restrict__ p2trans,
    const float* __restrict__ pos3, const float* __restrict__ h1,
    float* __restrict__ hin2) {
    __shared__ float4 sp[1024];
    __shared__ float4 st[1024];
    const int tid = threadIdx.x;
    const int t = blockIdx.x * 256 + tid;
    const float py0 = pos3[t * 3 + 0], py1 = pos3[t * 3 + 1], py2 = pos3[t * 3 + 2];
    const float s = 1.0f + SCALE_C * (py1 / SPAN_C);
    const float tx0 = (py0 - TAN30_C * py1) * s, tx1 = py1 * s, tx2 = py2 * s;
    float dP0 = 1e30f, dP1 = 1e30f, dP2 = 1e30f; int iP0 = 0, iP1 = 0, iP2 = 0;
    float dT0 = 1e30f, dT1 = 1e30f, dT2 = 1e30f; int iT0 = 0, iT1 = 0, iT2 = 0;
    for (int base = 0; base < N2V; base += 1024) {
        __syncthreads();
        for (int i = tid; i < 1024; i += 256) {
            sp[i] = ((const float4*)p2plain)[base + i];
            st[i] = ((const float4*)p2trans)[base + i];
        }
        __syncthreads();
        for (int j = 0; j < 1024; ++j) {
            const int idx = base + j;
            float4 q = sp[j];
            float dx = py0 - q.x, dy = py1 - q.y, dz = py2 - q.z;
            float d2 = dx * dx + dy * dy + dz * dz;
            if (d2 < dP2) {
                if (d2 < dP1) {
                    dP2 = dP1; iP2 = iP1;
                    if (d2 < dP0) { dP1 = dP0; iP1 = iP0; dP0 = d2; iP0 = idx; }
                    else { dP1 = d2; iP1 = idx; }
                } else { dP2 = d2; iP2 = idx; }
            }
            q = st[j];
            dx = tx0 - q.x; dy = tx1 - q.y; dz = tx2 - q.z;
            d2 = dx * dx + dy * dy + dz * dz;
            if (d2 < dT2) {
                if (d2 < dT1) {
                    dT2 = dT1; iT2 = iT1;
                    if (d2 < dT0) { dT1 = dT0; iT1 = iT0; dT0 = d2; iT0 = idx; }
                    else { dT1 = d2; iT1 = idx; }
                } else { dT2 = d2; iT2 = idx; }
            }
        }
    }
    const float wP0 = 1.f / (dP0 + 1e-16f), wP1 = 1.f / (dP1 + 1e-16f), wP2 = 1.f / (dP2 + 1e-16f);
    const float wT0 = 1.f / (dT0 + 1e-16f), wT1 = 1.f / (dT1 + 1e-16f), wT2 = 1.f / (dT2 + 1e-16f);
    const float rsP = 1.f / (wP0 + wP1 + wP2);
    const float rsT = 1.f / (wT0 + wT1 + wT2);
    const bool useT = (py1 < SPAN_C);
    const float* hp0 = h1 + (size_t)iP0 * 64;
    const float* hp1 = h1 + (size_t)iP1 * 64;
    const float* hp2 = h1 + (size_t)iP2 * 64;
    const float* ht0 = h1 + (size_t)iT0 * 64;
    const float* ht1 = h1 + (size_t)iT1 * 64;
    const float* ht2 = h1 + (size_t)iT2 * 64;
    for (int f = 0; f < 64; ++f) {
        float vP = (wP0 * hp0[f] + wP1 * hp1[f] + wP2 * hp2[f]) * rsP;
        float vT = (wT0 * ht0[f] + wT1 * ht1[f] + wT2 * ht2[f]) * rsT;
        hin2[(size_t)t * 64 + f] = useT ? vT : vP;
    }
}

// ---------------------------------------------------------------------------
extern "C" void kernel_launch(void* const* d_in, const int* in_sizes, int n_in,
                              void* d_out, int out_size, void* d_ws, size_t ws_size,
                              hipStream_t stream) {
    (void)n_in; (void)out_size; (void)ws_size;
    const float* x = (const float*)d_in[0];
    const float* pos2 = (const float*)d_in[1];
    const float* pos3 = (const float*)d_in[2];

    // logical: 0 mw0,1 mb0,2 mws,3 mbs,4 mlw,5 mlb,6 sw0,7 sb0,8 sws,9 sbs,10 lw,11 lb
    static const int MAP_SORT[12] = {6, 2, 7, 3, 5, 4, 10, 8, 11, 9, 1, 0};
    const bool sortedOrder = (in_sizes[3] != HID1 * 5);
    const float* P1[12];
    const float* P2[12];
    for (int i = 0; i < 12; ++i) {
        int j = sortedOrder ? MAP_SORT[i] : i;
        P1[i] = (const float*)d_in[3 + j];
        P2[i] = (const float*)d_in[15 + j];
    }

    char* wp_ = (char*)d_ws;
    auto carve = [&](size_t bytes) -> float* {
        float* p = (float*)wp_;
        wp_ += (bytes + 255) & ~(size_t)255;
        return p;
    };
    float* mm      = carve(12 * 4);
    float* pos2p   = carve((size_t)N2V * 4 * 4);
    float* pos3p   = carve((size_t)N3V * 4 * 4);
    float* xp      = carve((size_t)N2V * 8 * 4);
    float* p2plain = carve((size_t)N2V * 4 * 4);
    float* p2trans = carve((size_t)N2V * 4 * 4);
    float* bufA1   = carve((size_t)N2V * PAD1 * 4);
    float* bufB1   = carve((size_t)N2V * PAD1 * 4);
    float* alphas1 = carve((size_t)N2V * 4 * 4);
    float* h1      = carve((size_t)N2V * 64 * 4);
    float* hin2    = carve((size_t)N3V * 64 * 4);
    float* bufA2   = carve((size_t)N3V * PAD2 * 4);
    float* bufB2   = carve((size_t)N3V * PAD2 * 4);
    float* alphas2 = carve((size_t)N3V * 4 * 4);
    // packed weights/biases, stage 1
    float* w_mw0 = carve(MP1 * 8 * 4);            float* b_mb0 = carve(MP1 * 4);
    float* w_mws = carve(4ull * MP1 * PAD1 * 4);  float* b_mbs = carve(4 * MP1 * 4);
    float* w_mlw = carve(16 * PAD1 * 4);          float* b_mlb = carve(16 * 4);
    float* w_sw0 = carve(MP1 * 4 * 4);            float* b_sb0 = carve(MP1 * 4);
    float* w_sws = carve(4ull * MP1 * PAD1 * 4);  float* b_sbs = carve(4 * MP1 * 4);
    float* w_lw  = carve(64 * PAD1 * 4);          float* b_lb  = carve(64 * 4);
    // packed weights/biases, stage 2
    float* w2_mw0 = carve(16 * 64 * 4);           float* b2_mb0 = carve(16 * 4);
    float* w2_mws = carve(4 * 16 * PAD2 * 4);     float* b2_mbs = carve(4 * 16 * 4);
    float* w2_mlw = carve(16 * PAD2 * 4);         float* b2_mlb = carve(16 * 4);
    float* w2_sw0 = carve(16 * 4 * 4);            float* b2_sb0 = carve(16 * 4);
    float* w2_sws = carve(4 * 16 * PAD2 * 4);     float* b2_sbs = carve(4 * 16 * 4);
    float* w2_lw  = carve(16 * PAD2 * 4);         float* b2_lb  = carve(16 * 4);

    auto pack = [&](const float* W, const float* B, int M, int K, int Kp, int Mp,
                    float* Wp, float* Bp) {
        pack_kernel<<<(Mp * Kp + 255) / 256, 256, 0, stream>>>(W, B, M, K, Kp, Mp, Wp, Bp);
    };
    auto layer = [&](const float* X, int ldx, const float* Wp, const float* Bp,
                     const float* al, int l, float* Y, int ldy, int Mout,
                     int use_sine, int N) {
        dim3 grid(N / 16, (Mout + 15) / 16);
        if (al)
            wmma_sine_layer<true, true><<<grid, 32, 0, stream>>>(X, ldx, Wp, Bp, al, l, Y, ldy, Mout, OMEGA_C);
        else if (use_sine)
            wmma_sine_layer<false, true><<<grid, 32, 0, stream>>>(X, ldx, Wp, Bp, nullptr, 0, Y, ldy, Mout, OMEGA_C);
        else
            wmma_sine_layer<false, false><<<grid, 32, 0, stream>>>(X, ldx, Wp, Bp, nullptr, 0, Y, ldy, Mout, OMEGA_C);
    };

    // ---- weight packing ----
    pack(P1[0], P1[1], HID1, 5, 8, MP1, w_mw0, b_mb0);
    for (int l = 0; l < 4; ++l)
        pack(P1[2] + (size_t)l * HID1 * HID1, P1[3] + l * HID1, HID1, HID1, PAD1, MP1,
             w_mws + (size_t)l * MP1 * PAD1, b_mbs + l * MP1);
    pack(P1[4], P1[5], 4, HID1, PAD1, 16, w_mlw, b_mlb);
    pack(P1[6], P1[7], HID1, 3, 4, MP1, w_sw0, b_sb0);
    for (int l = 0; l < 4; ++l)
        pack(P1[8] + (size_t)l * HID1 * HID1, P1[9] + l * HID1, HID1, HID1, PAD1, MP1,
             w_sws + (size_t)l * MP1 * PAD1, b_sbs + l * MP1);
    pack(P1[10], P1[11], 64, HID1, PAD1, 64, w_lw, b_lb);
    pack(P2[0], P2[1], HID2, 64, 64, 16, w2_mw0, b2_mb0);
    for (int l = 0; l < 4; ++l)
        pack(P2[2] + l * HID2 * HID2, P2[3] + l * HID2, HID2, HID2, PAD2, 16,
             w2_mws + l * 16 * PAD2, b2_mbs + l * 16);
    pack(P2[4], P2[5], 4, HID2, PAD2, 16, w2_mlw, b2_mlb);
    pack(P2[6], P2[7], HID2, 3, 4, 16, w2_sw0, b2_sb0);
    for (int l = 0; l < 4; ++l)
        pack(P2[8] + l * HID2 * HID2, P2[9] + l * HID2, HID2, HID2, PAD2, 16,
             w2_sws + l * 16 * PAD2, b2_sbs + l * 16);
    pack(P2[10], P2[11], 5, HID2, PAD2, 16, w2_lw, b2_lb);

    minmax_kernel<<<2, 256, 0, stream>>>(pos2, pos3, mm);
    prep_kernel<<<(N3V + 255) / 256, 256, 0, stream>>>(x, pos2, pos3, mm, xp, pos2p,
                                                       pos3p, p2plain, p2trans);

    // ---- stage 1: modulation MLP (4096 rows, width 129) ----
    float* bufs[2] = {bufA1, bufB1};
    layer(xp, 8, w_mw0, b_mb0, nullptr, 0, bufA1, PAD1, PAD1, 1, N2V);
    for (int l = 0; l < 4; ++l)
        layer(bufs[l & 1], PAD1, w_mws + (size_t)l * MP1 * PAD1, b_mbs + l * MP1,
              nullptr, 0, bufs[(l & 1) ^ 1], PAD1, PAD1, 1, N2V);
    layer(bufA1, PAD1, w_mlw, b_mlb, nullptr, 0, alphas1, 4, 4, 0, N2V);
    // ---- stage 1: modulated SIREN ----
    layer(pos2p, 4, w_sw0, b_sb0, nullptr, 0, bufB1, PAD1, PAD1, 1, N2V);
    for (int l = 0; l < 4; ++l)
        layer(bufs[(l & 1) ^ 1], PAD1, w_sws + (size_t)l * MP1 * PAD1, b_sbs + l * MP1,
              alphas1, l, bufs[l & 1], PAD1, PAD1, 1, N2V);
    layer(bufB1, PAD1, w_lw, b_lb, nullptr, 0, h1, 64, 64, 0, N2V);

    // ---- KNN interpolation 4096 -> 16384 ----
    knn_kernel<<<N3V / 256, 256, 0, stream>>>(p2plain, p2trans, pos3, h1, hin2);

    // ---- stage 2: modulation MLP (16384 rows, width 11) ----
    float* bufs2[2] = {bufA2, bufB2};
    layer(hin2, 64, w2_mw0, b2_mb0, nullptr, 0, bufA2, PAD2, PAD2, 1, N3V);
    for (int l = 0; l < 4; ++l)
        layer(bufs2[l & 1], PAD2, w2_mws + l * 16 * PAD2, b2_mbs + l * 16,
              nullptr, 0, bufs2[(l & 1) ^ 1], PAD2, PAD2, 1, N3V);
    layer(bufA2, PAD2, w2_mlw, b2_mlb, nullptr, 0, alphas2, 4, 4, 0, N3V);
    // ---- stage 2: modulated SIREN ----
    layer(pos3p, 4, w2_sw0, b2_sb0, nullptr, 0, bufB2, PAD2, PAD2, 1, N3V);
    for (int l = 0; l < 4; ++l)
        layer(bufs2[(l & 1) ^ 1], PAD2, w2_sws + l * 16 * PAD2, b2_sbs + l * 16,
              alphas2, l, bufs2[l & 1], PAD2, PAD2, 1, N3V);
    layer(bufB2, PAD2, w2_lw, b2_lb, nullptr, 0, (float*)d_out, 5, 5, 0, N3V);
}